// CRF_59880434041416
// MI455X (gfx1250) — compile-verified
//
#include <hip/hip_runtime.h>
#include <stdint.h>

typedef __attribute__((ext_vector_type(2))) float v2f;
typedef __attribute__((ext_vector_type(8))) float v8f;

#define SEQ_LEN 512
#define BATCH   1024
#define NTAGS   64
#define EPAD    68   // padded row stride (floats) for 16-row LDS tiles
#define TPAD    68   // padded row stride (floats) for expT^T LDS tile

// ---------------------------------------------------------------------------
// Forward-algorithm scan. One wave (32 threads) per block handles 16 batch
// rows for all 512 timesteps. Per step:
//   D[j,b] = sum_i expT[i,j] * X[i,b]        (16 WMMA_F32_16x16x4 per jt-tile)
//   rel[j,b] = e[t,b,j] + log(D[j,b])        (relative to running m[b])
//   r[b] = max_j rel; m += r; X = exp(rel-r) (rescale, relayout via LDS)
// Orientation: batch in lanes everywhere, so D-layout -> next B-operand is a
// cheap padded-LDS roundtrip.
// ---------------------------------------------------------------------------
__global__ __launch_bounds__(32)
void crf_forward_scan(const float* __restrict__ em,
                      const uint8_t* __restrict__ mask,
                      const float* __restrict__ startT,
                      const float* __restrict__ endT,
                      const float* __restrict__ trans,
                      float* __restrict__ logZ)
{
    __shared__ float sT[NTAGS * TPAD];   // sT[j*TPAD + i] = exp(trans[i*64+j])  (= expT^T)
    __shared__ float sE[16 * EPAD];      // emissions tile for current t, b-major
    __shared__ float sX[16 * EPAD];      // rescaled-alpha relayout scratch, b-major
    __shared__ float sEnd[NTAGS];        // exp(end_transitions)

    const int lane = threadIdx.x;        // 0..31 (wave32)
    const int half = lane >> 4;          // 0 / 1
    const int bl   = lane & 15;          // batch row within tile
    const int b0   = blockIdx.x * 16;

    // ---- one-time LDS init ----
    for (int idx = lane; idx < NTAGS * NTAGS; idx += 32) {
        int i = idx >> 6, j = idx & 63;
        sT[j * TPAD + i] = __expf(trans[idx]);   // transpose while storing
    }
    for (int idx = lane; idx < NTAGS; idx += 32)
        sEnd[idx] = __expf(endT[idx]);

    // ---- stage emissions for t=0, prefetch t=1 ----
    float4 ereg[8];
    {
        const float4* gp = (const float4*)(em + (size_t)b0 * NTAGS);
        #pragma unroll
        for (int k = 0; k < 8; ++k) ereg[k] = gp[k * 32 + lane];
        #pragma unroll
        for (int k = 0; k < 8; ++k) {
            int f = k * 32 + lane;                       // 16B chunk index in 4KB tile
            *(float4*)&sE[(f >> 4) * EPAD + (f & 15) * 4] = ereg[k];
        }
        const float4* gp1 = (const float4*)(em + ((size_t)1 * BATCH + b0) * NTAGS);
        #pragma unroll
        for (int k = 0; k < 8; ++k) ereg[k] = gp1[k * 32 + lane];
    }
    __syncthreads();

    // ---- alpha0 = startT + e[0], built directly in B-operand layout ----
    // B layout (4x16 f32): chunk c, comp w, half h holds (i = 4c+w+2h, b = bl)
    v2f  Xv[16];
    float m;
    {
        float mx = -3.0e38f;
        #pragma unroll
        for (int c = 0; c < 16; ++c) {
            int i0 = 4 * c + 2 * half;
            float a0 = startT[i0]     + sE[bl * EPAD + i0];
            float a1 = startT[i0 + 1] + sE[bl * EPAD + i0 + 1];
            Xv[c].x = a0; Xv[c].y = a1;
            mx = fmaxf(mx, fmaxf(a0, a1));
        }
        mx = fmaxf(mx, __shfl_xor(mx, 16, 32));   // combine both halves (same b)
        m  = mx;
        #pragma unroll
        for (int c = 0; c < 16; ++c) {
            Xv[c].x = __expf(Xv[c].x - m);
            Xv[c].y = __expf(Xv[c].y - m);
        }
    }

    // ---- scan over time ----
    #pragma clang loop unroll(disable)
    for (int t = 1; t < SEQ_LEN; ++t) {
        // sE <- prefetched E[t]; prefetch E[t+1]
        #pragma unroll
        for (int k = 0; k < 8; ++k) {
            int f = k * 32 + lane;
            *(float4*)&sE[(f >> 4) * EPAD + (f & 15) * 4] = ereg[k];
        }
        if (t + 1 < SEQ_LEN) {
            const float4* gp = (const float4*)(em + ((size_t)(t + 1) * BATCH + b0) * NTAGS);
            #pragma unroll
            for (int k = 0; k < 8; ++k) ereg[k] = gp[k * 32 + lane];
        }

        // GEMM: acc[jt][v] = D[j=16jt+v+8h, b=bl] ; 4 independent chains x 16 K-steps
        v8f acc[4];
        #pragma unroll
        for (int jt = 0; jt < 4; ++jt) {
            v8f a = {};
            #pragma unroll
            for (int c = 0; c < 16; ++c) {
                // A operand (16x4 f32): float2 = expT^T[j, 4c+2h .. +1]
                v2f av = *(const v2f*)&sT[(16 * jt + bl) * TPAD + 4 * c + 2 * half];
                a = __builtin_amdgcn_wmma_f32_16x16x4_f32(
                        /*neg_a=*/false, av, /*neg_b=*/false, Xv[c],
                        /*c_mod=*/(short)0, a, /*reuse_a=*/false, /*reuse_b=*/false);
            }
            acc[jt] = a;
        }
        __syncthreads();   // sE stores complete before finalize reads

        // finalize: rel = e + log(D); rowmax over tags; rescale
        const bool mk = mask[(size_t)t * BATCH + b0 + bl] != 0;
        float r = -3.0e38f;
        #pragma unroll
        for (int jt = 0; jt < 4; ++jt) {
            #pragma unroll
            for (int v = 0; v < 8; ++v) {
                int j = 16 * jt + v + 8 * half;
                float rel = sE[bl * EPAD + j] + __logf(acc[jt][v]);
                acc[jt][v] = rel;
                r = fmaxf(r, rel);
            }
        }
        r = fmaxf(r, __shfl_xor(r, 16, 32));   // max over all 64 tags for this b

        // relayout D-layout -> B-layout through padded LDS
        #pragma unroll
        for (int jt = 0; jt < 4; ++jt) {
            #pragma unroll
            for (int v = 0; v < 8; ++v) {
                int j = 16 * jt + v + 8 * half;
                sX[bl * EPAD + j] = __expf(acc[jt][v] - r);
            }
        }
        __syncthreads();

        #pragma unroll
        for (int c = 0; c < 16; ++c) {
            v2f xn = *(const v2f*)&sX[bl * EPAD + 4 * c + 2 * half];
            Xv[c].x = mk ? xn.x : Xv[c].x;     // mask==0 -> alpha (and m) unchanged
            Xv[c].y = mk ? xn.y : Xv[c].y;
        }
        m = mk ? (m + r) : m;
    }

    // ---- logZ[b] = m + log( sum_j X[j,b] * exp(endT[j]) ) ----
    float acce = 0.f;
    #pragma unroll
    for (int c = 0; c < 16; ++c) {
        int i0 = 4 * c + 2 * half;
        acce += Xv[c].x * sEnd[i0] + Xv[c].y * sEnd[i0 + 1];
    }
    acce += __shfl_xor(acce, 16, 32);
    if (lane < 16) logZ[b0 + lane] = m + __logf(acce);
}

// ---------------------------------------------------------------------------
// Numerator (gold-path score) + final reduction: one thread per batch element.
// ---------------------------------------------------------------------------
__global__ __launch_bounds__(256)
void crf_numer(const float* __restrict__ em,
               const int* __restrict__ tags,
               const uint8_t* __restrict__ mask,
               const float* __restrict__ startT,
               const float* __restrict__ endT,
               const float* __restrict__ trans,
               const float* __restrict__ logZ,
               float* __restrict__ out)
{
    int b = blockIdx.x * blockDim.x + threadIdx.x;
    if (b >= BATCH) return;

    int   prev = tags[b];
    float mf0  = mask[b] ? 1.f : 0.f;
    float acc  = startT[prev] + em[(size_t)b * NTAGS + prev] * mf0;   // e at t=0 (t<=L-2)
    int   cnt  = mask[b] ? 1 : 0;

    for (int t = 1; t < SEQ_LEN; ++t) {
        int   tc = tags[(size_t)t * BATCH + b];
        bool  mb = mask[(size_t)t * BATCH + b] != 0;
        float mf = mb ? 1.f : 0.f;
        cnt += mb ? 1 : 0;
        acc += trans[prev * NTAGS + tc] * mf;
        if (t < SEQ_LEN - 1)
            acc += em[((size_t)t * BATCH + b) * NTAGS + tc] * mf;     // e_scores[:-1]
        prev = tc;
    }
    int   last_idx = cnt - 1;
    int   lt  = tags[(size_t)last_idx * BATCH + b];
    float mfL = mask[(size_t)(SEQ_LEN - 1) * BATCH + b] ? 1.f : 0.f;
    acc += endT[lt] + em[((size_t)(SEQ_LEN - 1) * BATCH + b) * NTAGS + lt] * mfL;

    atomicAdd(out, acc - logZ[b]);
}

// ---------------------------------------------------------------------------
extern "C" void kernel_launch(void* const* d_in, const int* in_sizes, int n_in,
                              void* d_out, int out_size, void* d_ws, size_t ws_size,
                              hipStream_t stream)
{
    const float*   em     = (const float*)d_in[0];
    const int*     tags   = (const int*)d_in[1];
    const uint8_t* mask   = (const uint8_t*)d_in[2];
    const float*   startT = (const float*)d_in[3];
    const float*   endT   = (const float*)d_in[4];
    const float*   trans  = (const float*)d_in[5];
    float* logZ = (float*)d_ws;                       // 1024 floats of scratch

    hipMemsetAsync(d_out, 0, sizeof(float), stream);

    hipLaunchKernelGGL(crf_forward_scan, dim3(BATCH / 16), dim3(32), 0, stream,
                       em, mask, startT, endT, trans, logZ);
    hipLaunchKernelGGL(crf_numer, dim3(BATCH / 256), dim3(256), 0, stream,
                       em, tags, mask, startT, endT, trans, logZ, (float*)d_out);
    (void)in_sizes; (void)n_in; (void)out_size; (void)ws_size;
}